// GridExp_52956946759701
// MI455X (gfx1250) — compile-verified
//
#include <hip/hip_runtime.h>

#define NX 192
#define NY 192
#define NZ 192
#define NVOX (NX * NY * NZ)
#define NELT (NVOX * 3)
#define HALF (NVOX / 2)   // = (NX/2) * NY * NZ

// floor-mod for dimension 192 (handles negative coords from the wrap boundary)
__device__ __forceinline__ int wrap192(int i) {
    int r = i % 192;
    return r + ((r >> 31) & 192);
}

__device__ __forceinline__ float lerpf(float a, float b, float w) {
    return fmaf(w, b - a, a);   // a + w*(b-a)
}

// paired lerp on (x,y) channels -> v_pk_fma_f32
__device__ __forceinline__ float2 lerp2(float2 a, float2 b, float w) {
    return make_float2(fmaf(w, b.x - a.x, a.x), fmaf(w, b.y - a.y, a.y));
}

// d0 = velocity * 2^-8, streaming float4 copy with prefetch-ahead into GL2.
__global__ __launch_bounds__(256) void init_scale(const float* __restrict__ v,
                                                  float* __restrict__ d0) {
    int i = blockIdx.x * 256 + threadIdx.x;
    const float4* vi = reinterpret_cast<const float4*>(v);
    float4* di = reinterpret_cast<float4*>(d0);
    if (i < NELT / 4) {
        __builtin_prefetch(vi + i + 2048, 0, 3);  // global_prefetch_b8
        float4 a = vi[i];
        const float s = 0.00390625f;  // 2^-8
        a.x *= s; a.y *= s; a.z *= s; a.w *= s;
        di[i] = a;
    }
}

// Per-voxel gather context: 8 corner element-offsets (32-bit, saddr-relative),
// lerp weights, self displacement, grid position.
struct VoxCtx {
    int off[8];
    float wx, wy, wz;
    float dx, dy, dz;
    float gx, gy, gz;
    int base;
};

__device__ __forceinline__ VoxCtx make_ctx(const float* __restrict__ src,
                                           int x, int y, int z, int base) {
    VoxCtx c;
    c.base = base;
    c.dx = src[base + 0];
    c.dy = src[base + 1];
    c.dz = src[base + 2];
    c.gx = (float)x; c.gy = (float)y; c.gz = (float)z;

    float cx = c.gx + c.dx, cy = c.gy + c.dy, cz = c.gz + c.dz;
    float fx = floorf(cx), fy = floorf(cy), fz = floorf(cz);
    c.wx = cx - fx; c.wy = cy - fy; c.wz = cz - fz;

    int ix0 = wrap192((int)fx), iy0 = wrap192((int)fy), iz0 = wrap192((int)fz);
    int ix1 = ix0 + 1; if (ix1 == NX) ix1 = 0;
    int iy1 = iy0 + 1; if (iy1 == NY) iy1 = 0;
    int iz1 = iz0 + 1; if (iz1 == NZ) iz1 = 0;

    int X0 = ix0 * (NY * NZ * 3), X1 = ix1 * (NY * NZ * 3);
    int Y0 = iy0 * (NZ * 3),      Y1 = iy1 * (NZ * 3);
    int Z0 = iz0 * 3,             Z1 = iz1 * 3;

    // bit0 = z, bit1 = y, bit2 = x
    c.off[0] = X0 + Y0 + Z0;  c.off[1] = X0 + Y0 + Z1;
    c.off[2] = X0 + Y1 + Z0;  c.off[3] = X0 + Y1 + Z1;
    c.off[4] = X1 + Y0 + Z0;  c.off[5] = X1 + Y0 + Z1;
    c.off[6] = X1 + Y1 + Z0;  c.off[7] = X1 + Y1 + Z1;
    return c;
}

// Trilinear reduction (lerp z, then y, then x) with packed (x,y)-channel math.
template <bool FINAL>
__device__ __forceinline__ void reduce_store(const VoxCtx& c, const float v[24],
                                             float* __restrict__ dst) {
    float2 p[8]; float q[8];
#pragma unroll
    for (int k = 0; k < 8; ++k) {
        p[k] = make_float2(v[3 * k + 0], v[3 * k + 1]);
        q[k] = v[3 * k + 2];
    }

    float2 c00 = lerp2(p[0], p[1], c.wz);  float z00 = lerpf(q[0], q[1], c.wz);
    float2 c01 = lerp2(p[2], p[3], c.wz);  float z01 = lerpf(q[2], q[3], c.wz);
    float2 c10 = lerp2(p[4], p[5], c.wz);  float z10 = lerpf(q[4], q[5], c.wz);
    float2 c11 = lerp2(p[6], p[7], c.wz);  float z11 = lerpf(q[6], q[7], c.wz);

    float2 c0 = lerp2(c00, c01, c.wy);     float z0 = lerpf(z00, z01, c.wy);
    float2 c1 = lerp2(c10, c11, c.wy);     float z1 = lerpf(z10, z11, c.wy);

    float2 s = lerp2(c0, c1, c.wx);        float sz = lerpf(z0, z1, c.wx);

    float rx = c.dx + s.x, ry = c.dy + s.y, rz = c.dz + sz;

    if (FINAL) {
        // out = id + d ; stream past the caches (never re-read)
        __builtin_nontemporal_store(rx + c.gx, dst + c.base + 0);
        __builtin_nontemporal_store(ry + c.gy, dst + c.base + 1);
        __builtin_nontemporal_store(rz + c.gz, dst + c.base + 2);
    } else {
        // intermediate field: RT so it stays resident in the 192MB L2
        dst[c.base + 0] = rx;
        dst[c.base + 1] = ry;
        dst[c.base + 2] = rz;
    }
}

// One scaling-and-squaring step, 2 voxels per thread: voxel i = (x,y,z) and
// voxel i+HALF = (x+96,y,z) — one div/mod chain serves both, and each lane
// keeps 16 independent b96 gathers in flight in one memory clause.
template <bool FINAL>
__global__ __launch_bounds__(256) void sq_step(const float* __restrict__ src,
                                               float* __restrict__ dst) {
    int i0 = blockIdx.x * 256 + threadIdx.x;
    if (i0 >= HALF) return;

    int z = i0 % NZ;
    int t = i0 / NZ;
    int y = t % NY;
    int x = t / NY;                 // x in [0, 96)

    VoxCtx A = make_ctx(src, x,            y, z, i0 * 3);
    VoxCtx B = make_ctx(src, x + (NX / 2), y, z, (i0 + HALF) * 3);

    float va[24], vb[24];
#pragma unroll
    for (int k = 0; k < 8; ++k) {
        int o = A.off[k];
        va[3 * k + 0] = src[o + 0];
        va[3 * k + 1] = src[o + 1];
        va[3 * k + 2] = src[o + 2];
    }
#pragma unroll
    for (int k = 0; k < 8; ++k) {
        int o = B.off[k];
        vb[3 * k + 0] = src[o + 0];
        vb[3 * k + 1] = src[o + 1];
        vb[3 * k + 2] = src[o + 2];
    }

    reduce_store<FINAL>(A, va, dst);
    reduce_store<FINAL>(B, vb, dst);
}

extern "C" void kernel_launch(void* const* d_in, const int* in_sizes, int n_in,
                              void* d_out, int out_size, void* d_ws, size_t ws_size,
                              hipStream_t stream) {
    (void)in_sizes; (void)n_in; (void)out_size; (void)ws_size;
    const float* vel = (const float*)d_in[0];
    float* out = (float*)d_out;   // ping-pong buffer B (81 MB)
    float* wsA = (float*)d_ws;    // ping-pong buffer A (81 MB) — both L2-resident

    dim3 blk(256);

    // d0 = v / 2^8  -> out
    init_scale<<<(NELT / 4) / 256, blk, 0, stream>>>(vel, out);

    const int g = HALF / 256;  // 13824 blocks, exact (2 voxels/thread)
    float* src = out;
    float* dst = wsA;
    // steps 1..7: L2-resident ping-pong
    for (int s = 0; s < 7; ++s) {
        sq_step<false><<<g, blk, 0, stream>>>(src, dst);
        float* tmp = src; src = dst; dst = tmp;
    }
    // step 8 fused with "+grid": ws -> out, non-temporal final store
    sq_step<true><<<g, blk, 0, stream>>>(src, out);
}